// EdgeAwareLoss_17205638988420
// MI455X (gfx1250) — compile-verified
//
#include <hip/hip_runtime.h>
#include <hip/hip_bf16.h>
#include <math.h>

// ---------------------------------------------------------------------------
// EdgeAwareLoss, fused single pass.
//   pred,target: [32,3,512,512] fp32 (planar NCHW)
//   out: 1 fp32 scalar
// Bandwidth-bound: 201 MB @ 23.3 TB/s ~= 8.6 us floor.
// ---------------------------------------------------------------------------

typedef float v2f __attribute__((ext_vector_type(2)));
typedef float v8f __attribute__((ext_vector_type(8)));

// Raw v_sqrt_f32 (<=1 ulp) instead of the correctly-rounded libm expansion:
// saves ~8-10 VALU ops per call in the hot loop.
__device__ __forceinline__ float fast_sqrtf(float x) {
#if defined(__gfx1250__) && __has_builtin(__builtin_amdgcn_sqrtf)
  return __builtin_amdgcn_sqrtf(x);
#else
  return sqrtf(x);
#endif
}

// Exact 32-lane sum using V_WMMA_F32_16X16X4_F32 (layout-safe construction):
//   A vgpr0 = p[lane], A vgpr1 = 0  -> A(m,0)=p[m], A(m,2)=p[m+16], others 0
//   B = all ones (layout irrelevant), C = 0
//   D(m,n) = p[m] + p[m+16]  (independent of n)
//   t[lane] = sum of 8 D vgprs = sum_{m in half} (p[m]+p[m+16])
//   t[lane] + t[lane^16] = sum over all 32 lanes.
__device__ __forceinline__ float wave_sum32(float v) {
#if defined(__gfx1250__) && __has_builtin(__builtin_amdgcn_wmma_f32_16x16x4_f32)
  v2f a; a[0] = v;    a[1] = 0.0f;
  v2f b; b[0] = 1.0f; b[1] = 1.0f;
  v8f c = {0.f, 0.f, 0.f, 0.f, 0.f, 0.f, 0.f, 0.f};
  v8f d = __builtin_amdgcn_wmma_f32_16x16x4_f32(false, a, false, b,
                                                (short)0, c, false, false);
  float t = ((d[0] + d[1]) + (d[2] + d[3])) + ((d[4] + d[5]) + (d[6] + d[7]));
  t += __shfl_xor(t, 16, 32);
  return t;
#else
  for (int off = 16; off; off >>= 1) v += __shfl_xor(v, off, 32);
  return v;
#endif
}

namespace {
constexpr int W          = 512;
constexpr int H          = 512;
constexpr int CH_STRIDE  = W * H;          // 262144
constexpr int IMG_STRIDE = 3 * CH_STRIDE;
constexpr int TILE       = 64;             // output tile (64x64)
constexpr int HA         = TILE + 2;       // halo dim = 66
constexpr int HELEMS     = HA * HA;        // 4356
constexpr int NWG        = 8 * 8 * 32;     // 2048 partial triples
}

__global__ __launch_bounds__(256) void edge_loss_tile(
    const float* __restrict__ pred, const float* __restrict__ targ,
    float* __restrict__ ws) {
  __shared__ float gp[HELEMS];
  __shared__ float gt[HELEMS];
  __shared__ float red[3][8];

  const int tid = threadIdx.x;
  const int hx0 = blockIdx.x * TILE - 1;
  const int hy0 = blockIdx.y * TILE - 1;
  const float* __restrict__ pB = pred + (size_t)blockIdx.z * IMG_STRIDE;
  const float* __restrict__ tB = targ + (size_t)blockIdx.z * IMG_STRIDE;

  // ---- Phase 1: gray halo tiles into LDS (zero pad outside image) ----
  for (int i = tid; i < HELEMS; i += 256) {
    const int r  = i / HA;
    const int cc = i - r * HA;
    const int y = hy0 + r;
    const int x = hx0 + cc;
    float g1 = 0.0f, g2 = 0.0f;
    if ((unsigned)y < (unsigned)H && (unsigned)x < (unsigned)W) {
      const int off = (y << 9) + x;
      g1 = fmaf(0.299f, pB[off],
           fmaf(0.587f, pB[off + CH_STRIDE], 0.114f * pB[off + 2 * CH_STRIDE]));
      g2 = fmaf(0.299f, tB[off],
           fmaf(0.587f, tB[off + CH_STRIDE], 0.114f * tB[off + 2 * CH_STRIDE]));
    }
    gp[i] = g1;
    gt[i] = g2;
  }
  __syncthreads();

  // ---- Phase 2: 3x3 stencils, sliding 3-row window, 16 rows per thread ----
  const int lx = tid & 63;   // output column in tile
  const int ly = tid >> 6;   // row group (0..3), rows ly*16 .. ly*16+15
  const int base = (ly * 16) * HA + lx;  // top-left of first 3x3 window

  float pa0 = gp[base],      pa1 = gp[base + 1],      pa2 = gp[base + 2];
  float ta0 = gt[base],      ta1 = gt[base + 1],      ta2 = gt[base + 2];
  float pb0 = gp[base + HA], pb1 = gp[base + HA + 1], pb2 = gp[base + HA + 2];
  float tb0 = gt[base + HA], tb1 = gt[base + HA + 1], tb2 = gt[base + HA + 2];

  float accS = 0.0f, accL = 0.0f, accH = 0.0f;

#pragma unroll 4
  for (int i = 0; i < 16; ++i) {
    const int ro = base + (i + 2) * HA;
    const float pc0 = gp[ro], pc1 = gp[ro + 1], pc2 = gp[ro + 2];
    const float tc0 = gt[ro], tc1 = gt[ro + 1], tc2 = gt[ro + 2];

    // pred edges
    float pex  = (pa2 - pa0) + 2.0f * (pb2 - pb0) + (pc2 - pc0);
    float pey  = (pc0 + 2.0f * pc1 + pc2) - (pa0 + 2.0f * pa1 + pa2);
    float psb  = fast_sqrtf(fmaf(pex, pex, fmaf(pey, pey, 1e-8f)));
    float plap = fabsf(pa1 + pb0 + pb2 + pc1 - 4.0f * pb1);
    float ps8  = (pa0 + pa1 + pa2) + (pb0 + pb2) + (pc0 + pc1 + pc2);
    float php  = fabsf(fmaf(8.0f, pb1, -ps8));

    // target edges
    float tex  = (ta2 - ta0) + 2.0f * (tb2 - tb0) + (tc2 - tc0);
    float tey  = (tc0 + 2.0f * tc1 + tc2) - (ta0 + 2.0f * ta1 + ta2);
    float tsb  = fast_sqrtf(fmaf(tex, tex, fmaf(tey, tey, 1e-8f)));
    float tlap = fabsf(ta1 + tb0 + tb2 + tc1 - 4.0f * tb1);
    float ts8  = (ta0 + ta1 + ta2) + (tb0 + tb2) + (tc0 + tc1 + tc2);
    float thp  = fabsf(fmaf(8.0f, tb1, -ts8));

    const float dS = psb - tsb;   accS = fmaf(dS, dS, accS);
    const float dL = plap - tlap; accL = fmaf(dL, dL, accL);
    const float dH = php - thp;   accH = fmaf(dH, dH, accH);

    // slide window down one row
    pa0 = pb0; pa1 = pb1; pa2 = pb2;  pb0 = pc0; pb1 = pc1; pb2 = pc2;
    ta0 = tb0; ta1 = tb1; ta2 = tb2;  tb0 = tc0; tb1 = tc1; tb2 = tc2;
  }

  // ---- Phase 3: hierarchical reduction (WMMA wave sum -> LDS -> ws) ----
  accS = wave_sum32(accS);
  accL = wave_sum32(accL);
  accH = wave_sum32(accH);

  const int lane = tid & 31;
  const int wv   = tid >> 5;
  if (lane == 0) { red[0][wv] = accS; red[1][wv] = accL; red[2][wv] = accH; }
  __syncthreads();
  if (tid == 0) {
    float s0 = 0.f, s1 = 0.f, s2 = 0.f;
#pragma unroll
    for (int w = 0; w < 8; ++w) { s0 += red[0][w]; s1 += red[1][w]; s2 += red[2][w]; }
    const int wg = blockIdx.x + (blockIdx.y << 3) + (blockIdx.z << 6);
    ws[wg * 3 + 0] = s0;
    ws[wg * 3 + 1] = s1;
    ws[wg * 3 + 2] = s2;
  }
}

__global__ __launch_bounds__(256) void edge_loss_final(
    const float* __restrict__ ws, float* __restrict__ out) {
  __shared__ float red[3][8];
  const int tid = threadIdx.x;
  float s0 = 0.f, s1 = 0.f, s2 = 0.f;
  for (int i = tid; i < NWG; i += 256) {
    s0 += ws[3 * i + 0];
    s1 += ws[3 * i + 1];
    s2 += ws[3 * i + 2];
  }
  s0 = wave_sum32(s0);
  s1 = wave_sum32(s1);
  s2 = wave_sum32(s2);

  const int lane = tid & 31;
  const int wv   = tid >> 5;
  if (lane == 0) { red[0][wv] = s0; red[1][wv] = s1; red[2][wv] = s2; }
  __syncthreads();
  if (tid == 0) {
    float S = 0.f, L = 0.f, Hs = 0.f;
#pragma unroll
    for (int w = 0; w < 8; ++w) { S += red[0][w]; L += red[1][w]; Hs += red[2][w]; }
    const float inv = 1.0f / 8388608.0f;  // 32*1*512*512 elements per mean
    const float ms = S * inv, ml = L * inv, mh = Hs * inv;
    float loss = fmaf(2.0f, mh, fmaf(1.5f, ml, ms));
    if (!isfinite(loss)) loss = 0.0f;
    loss = fminf(fmaxf(loss, 0.0f), 10.0f);
    out[0] = loss;
  }
}

extern "C" void kernel_launch(void* const* d_in, const int* in_sizes, int n_in,
                              void* d_out, int out_size, void* d_ws, size_t ws_size,
                              hipStream_t stream) {
  (void)in_sizes; (void)n_in; (void)out_size; (void)ws_size;
  const float* pred = (const float*)d_in[0];
  const float* targ = (const float*)d_in[1];
  float* ws = (float*)d_ws;   // 2048 * 3 floats = 24 KB of partial sums

  dim3 grid(8, 8, 32);        // 64x64 tiles over 512x512, 32 batch images
  edge_loss_tile<<<grid, 256, 0, stream>>>(pred, targ, ws);
  edge_loss_final<<<1, 256, 0, stream>>>(ws, (float*)d_out);
}